// GaussianProjection3D_86912958202322
// MI455X (gfx1250) — compile-verified
//
#include <hip/hip_runtime.h>

// GaussianProjection3D for MI455X (gfx1250, wave32).
// num = w_out @ M + b_out (x) S,  M[33,64] = [feat;1s][33,V] x WfullT[V,64]
// Main GEMM done with V_WMMA_F32_16X16X4_F32 (FP32, K stepped 4 voxels).

typedef float v2f __attribute__((ext_vector_type(2)));
typedef float v8f __attribute__((ext_vector_type(8)));

#define CBATCH 2
#define CCH    32
#define CV     (64*64*64)      // 262144 voxels
#define NBLK   128             // blocks per batch
#define VPB    (CV / NBLK)     // 2048 voxels per block
#define NGRP   (VPB / 256)     // 8 groups of 32 voxels per wave
#define MROWS  33              // 32 channels + ones row (for S)
#define MGSZ   (MROWS * 64)    // 2112 floats per batch in workspace

__device__ __forceinline__ float fast_tanh(float x) {
    float e = __expf(2.0f * x);
    return (e - 1.0f) * __builtin_amdgcn_rcpf(e + 1.0f);
}

__global__ void gp3d_init(float* __restrict__ p, int n) {
    int i = blockIdx.x * blockDim.x + threadIdx.x;
    if (i < n) p[i] = 0.0f;
}

__global__ __launch_bounds__(256)
void gp3d_main(const float* __restrict__ feat,
               const float* __restrict__ w_param,
               const float* __restrict__ b_param,
               float* __restrict__ macc_g)
{
    __shared__ __align__(16) float macc[MGSZ];         // block-level M partial
    __shared__ __align__(16) float feat_s[8][32 * 33]; // per-wave feat tile, stride 33
    __shared__ __align__(16) float kern_s[8][32 * 20]; // per-voxel {kx[4], kyz[16]}

    const int tid  = threadIdx.x;
    const int lane = tid & 31;
    const int wv   = tid >> 5;
    const int hi   = lane >> 4;   // lane half (K select in WMMA layouts)
    const int lo   = lane & 15;   // M (A tiles) / N (B,C,D tiles)
    const int b    = blockIdx.y;

    for (int i = tid; i < MGSZ; i += 256) macc[i] = 0.0f;
    __syncthreads();

    const float* featb = feat + (size_t)b * CCH * CV;
    float* fs = feat_s[wv];
    float* ks = kern_s[wv];

    // Ones-row A tile (row 0 of m=2 block == all ones -> D row 0 = S segment)
    const float onesv = (lo == 0) ? 1.0f : 0.0f;
    v2f a2; a2.x = onesv; a2.y = onesv;

    v8f acc[3][4];
    #pragma unroll
    for (int m = 0; m < 3; ++m)
        #pragma unroll
        for (int gq = 0; gq < 4; ++gq)
            #pragma unroll
            for (int r = 0; r < 8; ++r) acc[m][gq][r] = 0.0f;

    for (int grp = 0; grp < NGRP; ++grp) {
        const int vbase = blockIdx.x * VPB + wv * (NGRP * 32) + grp * 32;
        const int vv    = vbase + lane;   // this lane's voxel

        // ---- stage feat (coalesced) + per-voxel pars matvec ----
        float p0 = 0, p1 = 0, p2 = 0, p3 = 0, p4 = 0, p5 = 0;
        #pragma unroll
        for (int c = 0; c < CCH; ++c) {
            float f = featb[(size_t)c * CV + vv];
            fs[c * 33 + lane] = f;
            p0 = fmaf(w_param[0 * CCH + c], f, p0);
            p1 = fmaf(w_param[1 * CCH + c], f, p1);
            p2 = fmaf(w_param[2 * CCH + c], f, p2);
            p3 = fmaf(w_param[3 * CCH + c], f, p3);
            p4 = fmaf(w_param[4 * CCH + c], f, p4);
            p5 = fmaf(w_param[5 * CCH + c], f, p5);
        }
        p0 += b_param[0]; p1 += b_param[1]; p2 += b_param[2];
        p3 += b_param[3]; p4 += b_param[4]; p5 += b_param[5];

        const float mu0 = fast_tanh(p0), mu1 = fast_tanh(p1), mu2 = fast_tanh(p2);
        const float s0 = __expf(p3) + 0.001f;
        const float s1 = __expf(p4) + 0.001f;
        const float s2 = __expf(p5) + 0.001f;
        const float n0 = -0.5f * __builtin_amdgcn_rcpf(s0 * s0);
        const float n1 = -0.5f * __builtin_amdgcn_rcpf(s1 * s1);
        const float n2 = -0.5f * __builtin_amdgcn_rcpf(s2 * s2);

        const float x = (float)(vv >> 12)        * (2.0f / 63.0f) - 1.0f;
        const float y = (float)((vv >> 6) & 63)  * (2.0f / 63.0f) - 1.0f;
        const float z = (float)(vv & 63)         * (2.0f / 63.0f) - 1.0f;

        float kx[4], kyv[4], kzv[4];
        #pragma unroll
        for (int gi = 0; gi < 4; ++gi) {
            const float ctr = (float)gi * (2.0f / 3.0f) - 1.0f;
            float dx = x - ctr - mu0; kx[gi]  = __expf(dx * dx * n0);
            float dy = y - ctr - mu1; kyv[gi] = __expf(dy * dy * n1);
            float dz = z - ctr - mu2; kzv[gi] = __expf(dz * dz * n2);
        }

        float* kv = ks + lane * 20;                       // 80B stride, 16B aligned
        *(float4*)(kv) = make_float4(kx[0], kx[1], kx[2], kx[3]);
        #pragma unroll
        for (int j = 0; j < 4; ++j)
            *(float4*)(kv + 4 + 4 * j) =
                make_float4(kyv[j] * kzv[0], kyv[j] * kzv[1],
                            kyv[j] * kzv[2], kyv[j] * kzv[3]);

        // ---- WMMA contraction over this 32-voxel group (K steps of 4) ----
        // Wave-private LDS staging: per-wave DS ordering handled by DScnt waits.
        #pragma unroll
        for (int kstep = 0; kstep < 8; ++kstep) {
            const int vb = kstep * 4;
            const int vA = vb + 2 * hi;  // voxel of component .x (K = 2*hi + comp)

            // A tiles: 16x4 f32 layout (lane -> M=lo; VGPR comp + lane-half -> K)
            v2f A0, A1;
            A0.x = fs[lo * 33 + vA];            A0.y = fs[lo * 33 + vA + 1];
            A1.x = fs[(lo + 16) * 33 + vA];     A1.y = fs[(lo + 16) * 33 + vA + 1];

            // B tiles: 4x16, lane -> N=lo; element = kx[v][gq] * kyz[v][lo]
            const float* klo = ks + vA * 20;
            const float* khi = ks + (vA + 1) * 20;
            const float  kyz0 = klo[4 + lo];
            const float  kyz1 = khi[4 + lo];
            const float4 kx0  = *(const float4*)klo;
            const float4 kx1  = *(const float4*)khi;
            v2f Bt[4];
            Bt[0].x = kx0.x * kyz0; Bt[0].y = kx1.x * kyz1;
            Bt[1].x = kx0.y * kyz0; Bt[1].y = kx1.y * kyz1;
            Bt[2].x = kx0.z * kyz0; Bt[2].y = kx1.z * kyz1;
            Bt[3].x = kx0.w * kyz0; Bt[3].y = kx1.w * kyz1;

            #pragma unroll
            for (int gq = 0; gq < 4; ++gq) {
                acc[0][gq] = __builtin_amdgcn_wmma_f32_16x16x4_f32(
                    false, A0, false, Bt[gq], (short)0, acc[0][gq], false, false);
                acc[1][gq] = __builtin_amdgcn_wmma_f32_16x16x4_f32(
                    false, A1, false, Bt[gq], (short)0, acc[1][gq], false, false);
                acc[2][gq] = __builtin_amdgcn_wmma_f32_16x16x4_f32(
                    false, a2, false, Bt[gq], (short)0, acc[2][gq], false, false);
            }
        }
    }

    // ---- cross-wave reduction: ds_add_f32 into shared accumulator ----
    // C/D layout: VGPR r, lane half -> row = r + 8*hi; lane lo -> column.
    #pragma unroll
    for (int m = 0; m < 3; ++m)
        #pragma unroll
        for (int gq = 0; gq < 4; ++gq)
            #pragma unroll
            for (int r = 0; r < 8; ++r) {
                const int row = 16 * m + r + 8 * hi;
                if (row < MROWS)
                    atomicAdd(&macc[row * 64 + gq * 16 + lo], acc[m][gq][r]);
            }
    __syncthreads();

    // ---- flush block partial to global workspace ----
    float* dst = macc_g + (size_t)b * MGSZ;
    for (int i = tid; i < MGSZ; i += 256)
        atomicAdd(&dst[i], macc[i]);
}

__global__ void gp3d_fin(const float* __restrict__ w_out,
                         const float* __restrict__ b_out,
                         const float* __restrict__ macc_g,
                         float* __restrict__ out)
{
    int idx = blockIdx.x * blockDim.x + threadIdx.x;
    if (idx >= CBATCH * 64 * 64) return;
    const int g  = idx & 63;
    const int co = (idx >> 6) & 63;
    const int b  = idx >> 12;
    const float* M = macc_g + (size_t)b * MGSZ;
    const float S = M[32 * 64 + g];
    float num = b_out[co] * S;
    #pragma unroll
    for (int c = 0; c < CCH; ++c)
        num = fmaf(w_out[co * CCH + c], M[c * 64 + g], num);
    out[idx] = num / (S + 1e-6f);
}

extern "C" void kernel_launch(void* const* d_in, const int* in_sizes, int n_in,
                              void* d_out, int out_size, void* d_ws, size_t ws_size,
                              hipStream_t stream) {
    const float* feat    = (const float*)d_in[0];
    const float* w_param = (const float*)d_in[1];
    const float* b_param = (const float*)d_in[2];
    const float* w_out   = (const float*)d_in[3];
    const float* b_out   = (const float*)d_in[4];
    float* out  = (float*)d_out;
    float* macc = (float*)d_ws;   // needs CBATCH*MGSZ*4 = 16.9 KB

    const int ntot = CBATCH * MGSZ;
    gp3d_init<<<(ntot + 255) / 256, 256, 0, stream>>>(macc, ntot);
    gp3d_main<<<dim3(NBLK, CBATCH), 256, 0, stream>>>(feat, w_param, b_param, macc);
    gp3d_fin<<<(CBATCH * 64 * 64 + 255) / 256, 256, 0, stream>>>(w_out, b_out, macc, out);
}